// SampleLayer_45724221833750
// MI455X (gfx1250) — compile-verified
//
#include <hip/hip_runtime.h>

// Problem constants from the reference: B, L, D, S = (256, 200, 128, 10)
constexpr int kB   = 256;
constexpr int kL   = 200;
constexpr int kD   = 128;
constexpr int kS   = 10;
constexpr int kLM1 = kL - 1;      // 199
constexpr int kD4  = kD / 4;      // 32 float4 per row == one wave32, one float4/lane

typedef float v4f __attribute__((ext_vector_type(4)));

// ---------------------------------------------------------------------------
// pos_seq = x[:, 1:, :]  -- per-b contiguous copy of 199*128 floats.
// One float4 (16B) per thread; NT stores stream past L2.
// ---------------------------------------------------------------------------
__global__ __launch_bounds__(256) void pos_copy_kernel(
    const v4f* __restrict__ x4, v4f* __restrict__ out4, int total4, int xmax4) {
  int i = blockIdx.x * blockDim.x + threadIdx.x;
  if (i >= total4) return;

  constexpr int per_b = kLM1 * kD4;          // float4s per batch
  int b   = i / per_b;
  int rem = i - b * per_b;
  int src = b * (kL * kD4) + kD4 + rem;      // skip row 0 of each batch

  // Run-ahead prefetch of the streaming source (clamped in-bounds so the
  // global_prefetch_b8 stays on valid pages).
  int pf = src + 8 * 256;                    // 8 blocks ahead (~32 KB)
  if (pf >= xmax4) pf = xmax4 - 1;
  __builtin_prefetch(&x4[pf], 0, 0);

  v4f v = x4[src];
  __builtin_nontemporal_store(v, out4 + i);  // TH=NT: don't pollute L2
}

// ---------------------------------------------------------------------------
// neg_seq[b, j, s, :] = x[b, neg_idx[j, s], :]
// 32 consecutive lanes = one (b,j,s) row; neg_idx lookup is wave-uniform so
// force it scalar with readfirstlane. Reads of x hit L2 (x is 26 MB, L2 192 MB
// and NT output stores keep it resident); writes stream NT to HBM.
// ---------------------------------------------------------------------------
__global__ __launch_bounds__(256) void neg_gather_kernel(
    const v4f* __restrict__ x4, const int* __restrict__ neg_idx,
    v4f* __restrict__ out4, int total4) {
  int i = blockIdx.x * blockDim.x + threadIdx.x;
  if (i >= total4) return;

  int lane = i & (kD4 - 1);                  // 0..31, float4 within the row
  int row  = i >> 5;                         // flat (b, j, s) -- wave-uniform
  constexpr int rows_per_b = kLM1 * kS;      // 1990
  int b   = row / rows_per_b;
  int rem = row - b * rows_per_b;            // j*S + s == flat index into neg_idx

  // Wave-uniform gather index -> SGPR; the b128 load then uses saddr form.
  int src_l = __builtin_amdgcn_readfirstlane(neg_idx[rem]);

  v4f v = x4[(b * kL + src_l) * kD4 + lane];
  __builtin_nontemporal_store(v, out4 + i);  // TH=NT
}

// ---------------------------------------------------------------------------
// Launch: d_in[0] = x (float32, B*L*D), d_in[1] = neg_idx (int32, (L-1)*S)
// d_out = [pos_seq | neg_seq] concatenated flat, fp32.
// ---------------------------------------------------------------------------
extern "C" void kernel_launch(void* const* d_in, const int* in_sizes, int n_in,
                              void* d_out, int out_size, void* d_ws, size_t ws_size,
                              hipStream_t stream) {
  (void)in_sizes; (void)n_in; (void)out_size; (void)d_ws; (void)ws_size;

  const v4f* x4      = (const v4f*)d_in[0];
  const int* neg_idx = (const int*)d_in[1];

  float* out  = (float*)d_out;
  v4f*   pos4 = (v4f*)out;
  v4f*   neg4 = (v4f*)(out + (size_t)kB * kLM1 * kD);   // pos_seq elements first

  constexpr int pos_total4 = kB * kLM1 * kD4;           // 1,630,208 float4
  constexpr int neg_total4 = kB * kLM1 * kS * kD4;      // 16,302,080 float4
  constexpr int x_total4   = kB * kL * kD4;             // 1,638,400 float4

  constexpr int kBlk = 256;
  pos_copy_kernel<<<(pos_total4 + kBlk - 1) / kBlk, kBlk, 0, stream>>>(
      x4, pos4, pos_total4, x_total4);
  neg_gather_kernel<<<(neg_total4 + kBlk - 1) / kBlk, kBlk, 0, stream>>>(
      x4, neg_idx, neg4, neg_total4);
}